// QuantumFlowMatching_33466385170785
// MI455X (gfx1250) — compile-verified
//
#include <hip/hip_runtime.h>
#include <math.h>

typedef __attribute__((ext_vector_type(2))) float v2f;
typedef __attribute__((ext_vector_type(8))) float v8f;

#define NQ   16
#define QDIM 65536
#define BATCH 128
#define HID  192

static __device__ __forceinline__ float2 cmul(float2 a, float2 b) {
  return make_float2(a.x * b.x - a.y * b.y, a.x * b.y + a.y * b.x);
}

// Low 32 bits of a flat LDS pointer == LDS byte address (ISA 10.2 aperture rules).
static __device__ __forceinline__ unsigned lds_addr32(const void* p) {
  return (unsigned)(uintptr_t)p;
}

// ---------------------------------------------------------------------------
// init: interleave planar re/im into complex64-layout statevector (in d_out)
// ---------------------------------------------------------------------------
__global__ __launch_bounds__(256) void k_init(const float* __restrict__ re,
                                              const float* __restrict__ im,
                                              float2* __restrict__ psi) {
  size_t i = (size_t)blockIdx.x * 256 + threadIdx.x;
  psi[i] = make_float2(re[i], im[i]);
}

// ---------------------------------------------------------------------------
// timenet MLP + fused 16x16 group-unitary construction. One block per step.
// Group g covers wires 4g..4g+3 (statevector bits 15-4g .. 12-4g).
// G = C(4g+2,4g+3) C(4g+1,4g+2) C(4g,4g+1) * (U_{4g} (x) ... (x) U_{4g+3})
// Cross-group CNOTs are folded later as a column-xor (b^8) on the next G.
// ---------------------------------------------------------------------------
__global__ __launch_bounds__(HID) void k_net(
    const float* __restrict__ W1, const float* __restrict__ b1,
    const float* __restrict__ W2, const float* __restrict__ b2,
    const float* __restrict__ W3, const float* __restrict__ b3,
    float2* __restrict__ gm) {
  __shared__ float h1[HID], h2[HID], ang[48];
  __shared__ float2 U[NQ][2][2];
  const int tid = threadIdx.x;
  const int s = blockIdx.x;
  const float t = (float)s * 0.125f;

  { float x = W1[tid] * t + b1[tid];
    h1[tid] = x / (1.f + expf(-x)); }
  __syncthreads();
  { float a = b2[tid];
    for (int j = 0; j < HID; ++j) a += W2[tid * HID + j] * h1[j];
    h2[tid] = a / (1.f + expf(-a)); }
  __syncthreads();
  if (tid < 48) {
    float a = b3[tid];
    for (int j = 0; j < HID; ++j) a += W3[tid * HID + j] * h2[j];
    ang[tid] = a;
  }
  __syncthreads();
  if (tid < NQ) {
    float a = ang[tid*3+0]*0.5f, bh = ang[tid*3+1]*0.5f, c = ang[tid*3+2]*0.5f;
    float ca = cosf(a), sa = sinf(a), cb = cosf(bh), sb = sinf(bh);
    float cc = cosf(c), sc = sinf(c);
    // M = RY * RX
    float2 m00 = make_float2( cb*ca,  sb*sa);
    float2 m01 = make_float2(-sb*ca, -cb*sa);
    float2 m10 = make_float2( sb*ca, -cb*sa);
    float2 m11 = make_float2( cb*ca, -sb*sa);
    float2 zm = make_float2(cc, -sc);   // exp(-i c/2)
    float2 zp = make_float2(cc,  sc);   // exp(+i c/2)
    U[tid][0][0] = cmul(zm, m00); U[tid][0][1] = cmul(zm, m01);
    U[tid][1][0] = cmul(zp, m10); U[tid][1][1] = cmul(zp, m11);
  }
  __syncthreads();
  // 4 groups x 256 entries; CNOT chain = row permutation sigma1(sigma2(sigma3(a)))
  for (int e = tid; e < 1024; e += HID) {
    int g = e >> 8, ab = e & 255, a = ab >> 4, bcol = ab & 15;
    int aa = a;
    aa ^= ((aa >> 1) & 1);            // CNOT(local bit1 -> bit0)
    aa ^= ((aa >> 2) & 1) << 1;       // CNOT(local bit2 -> bit1)
    aa ^= ((aa >> 3) & 1) << 2;       // CNOT(local bit3 -> bit2)
    float2 v = make_float2(1.f, 0.f);
    for (int j = 0; j < 4; ++j) {
      int ai = (aa >> (3 - j)) & 1, bi = (bcol >> (3 - j)) & 1;
      v = cmul(v, U[4 * g + j][ai][bi]);
    }
    gm[(s * 4 + g) * 256 + a * 16 + bcol] = v;
  }
}

// ---------------------------------------------------------------------------
// One complex 16x16 gate application on a 16-column slice, via 16 f32 WMMAs.
// Assumed f32 operand layouts (wave32):
//   A (16x4): lane l, v[j] = A[l%16][j + 2*(l/16)]
//   B (4x16): lane l, v[j] = B[j + 2*(l/16)][l%16]
//   C/D     : lane l, v[j] = D[j + 8*(l/16)][l%16]
// xorm: column-xor on the matrix (folds a preceding conditional CNOT: M*X).
// ---------------------------------------------------------------------------
static __device__ __forceinline__ void wmma_group16(
    float2* __restrict__ buf, const float2* __restrict__ M,
    int colbase, int colstride, int rowstride, int xorm, int half, int lm) {
  v8f accr = {};
  v8f acci = {};
  for (int k = 0; k < 4; ++k) {
    const int j0 = 4 * k + 2 * half;
    float2 m0 = M[lm * 16 + (j0 ^ xorm)];
    float2 m1 = M[lm * 16 + ((j0 + 1) ^ xorm)];
    v2f Ar = {m0.x, m1.x};
    v2f Ai = {m0.y, m1.y};
    v2f An = {-m0.y, -m1.y};
    float2 p0 = buf[j0 * rowstride + colbase + lm * colstride];
    float2 p1 = buf[(j0 + 1) * rowstride + colbase + lm * colstride];
    v2f Br = {p0.x, p1.x};
    v2f Bi = {p0.y, p1.y};
    accr = __builtin_amdgcn_wmma_f32_16x16x4_f32(false, Ar, false, Br, (short)0, accr, false, false);
    accr = __builtin_amdgcn_wmma_f32_16x16x4_f32(false, An, false, Bi, (short)0, accr, false, false);
    acci = __builtin_amdgcn_wmma_f32_16x16x4_f32(false, Ar, false, Bi, (short)0, acci, false, false);
    acci = __builtin_amdgcn_wmma_f32_16x16x4_f32(false, Ai, false, Br, (short)0, acci, false, false);
  }
  for (int j = 0; j < 8; ++j) {
    int m = j + 8 * half;
    buf[m * rowstride + colbase + lm * colstride] = make_float2(accr[j], acci[j]);
  }
}

// ---------------------------------------------------------------------------
// H pass: G0 on bits 15-12 (wires 0-3), strided 16x16 tiles, in place in L2.
// Folds pending CNOT(15,0) (xor column by 8 when index bit0==1) and the
// pending per-row normalization scale of the previous step.
// One wave per tile; 8 waves / 256-thread block.
// ---------------------------------------------------------------------------
__global__ __launch_bounds__(256) void k_high(
    float2* __restrict__ psi, const float2* __restrict__ gm,
    const float* __restrict__ norms, int s, int pfold, int scale_step) {
  const int wave = threadIdx.x >> 5;
  const int lane = threadIdx.x & 31;
  const int half = lane >> 4, lm = lane & 15;
  const int tid = blockIdx.x * 8 + wave;       // 32768 tiles
  const int b = tid >> 8;
  const int rest = tid & 255;
  const int cb = rest >> 1, p = rest & 1;

  float inv = 1.0f;
  if (scale_step >= 0) {
    const float* np = norms + (scale_step * BATCH + b) * 16;
    float ss = 0.f;
    for (int i = 0; i < 16; ++i) ss += np[i];
    inv = 1.0f / (sqrtf(ss) + 1e-8f);
  }

  const float2* M = gm + (s * 4 + 0) * 256;
  const int xorm = (pfold && p) ? 8 : 0;
  const size_t base = ((size_t)b << 16) + (size_t)cb * 32 + 2 * lm + p;

  v8f accr = {};
  v8f acci = {};
  for (int k = 0; k < 4; ++k) {
    const int j0 = 4 * k + 2 * half;
    float2 m0 = M[lm * 16 + (j0 ^ xorm)];
    float2 m1 = M[lm * 16 + ((j0 + 1) ^ xorm)];
    v2f Ar = {m0.x, m1.x};
    v2f Ai = {m0.y, m1.y};
    v2f An = {-m0.y, -m1.y};
    float2 p0 = psi[base + ((size_t)j0 << 12)];
    float2 p1 = psi[base + ((size_t)(j0 + 1) << 12)];
    v2f Br = {p0.x, p1.x};
    v2f Bi = {p0.y, p1.y};
    accr = __builtin_amdgcn_wmma_f32_16x16x4_f32(false, Ar, false, Br, (short)0, accr, false, false);
    accr = __builtin_amdgcn_wmma_f32_16x16x4_f32(false, An, false, Bi, (short)0, accr, false, false);
    acci = __builtin_amdgcn_wmma_f32_16x16x4_f32(false, Ar, false, Bi, (short)0, acci, false, false);
    acci = __builtin_amdgcn_wmma_f32_16x16x4_f32(false, Ai, false, Br, (short)0, acci, false, false);
  }
  for (int j = 0; j < 8; ++j) {
    int m = j + 8 * half;
    psi[base + ((size_t)m << 12)] = make_float2(accr[j] * inv, acci[j] * inv);
  }
}

// ---------------------------------------------------------------------------
// L pass: one 4096-amplitude contiguous tile (bits 11-0) in LDS per block.
// Tile fill/drain uses CDNA5 async LDS DMA (ASYNCcnt-tracked), bypassing
// VGPRs: GLOBAL_LOAD_ASYNC_TO_LDS_B128 / GLOBAL_STORE_ASYNC_FROM_LDS_B128.
// Applies: [X on wire4 if bit12]  G1 (bits 11-8)
//          [X on wire8 if bit8]   G2 (bits 7-4)
//          [X on wire12 if bit4]  G3 (bits 3-0)
// (conditional Xs folded as matrix column-xor, per tile/column parity).
// Optionally emits deterministic per-(b,hi) |psi|^2 partial for normalization.
// ---------------------------------------------------------------------------
__global__ __launch_bounds__(256) void k_low(
    float2* __restrict__ psi, const float2* __restrict__ gm,
    float* __restrict__ norms, int s, int donorm) {
  __shared__ float2 tile[4096];
  __shared__ float red[256];
  const int b = blockIdx.x >> 4;
  const int hi = blockIdx.x & 15;
  const size_t gbase = ((size_t)b << 16) + ((size_t)hi << 12);

  // Async DMA: global -> LDS, 16B per lane per op; 512B/wave contiguous.
  for (int e = threadIdx.x * 2; e < 4096; e += 512) {
    unsigned la = lds_addr32(&tile[e]);
    const float2* ga = psi + gbase + e;
    asm volatile("global_load_async_to_lds_b128 %0, %1, off"
                 :: "v"(la), "v"(ga) : "memory");
  }
  asm volatile("s_wait_asynccnt 0" ::: "memory");
  __syncthreads();

  const int wave = threadIdx.x >> 5;
  const int lane = threadIdx.x & 31;
  const int half = lane >> 4, lm = lane & 15;

  // G1: rows = bits 11-8 (stride 256), cols = bits 7-0
  const float2* M1 = gm + (s * 4 + 1) * 256;
  const int x1 = (hi & 1) ? 8 : 0;   // CNOT(3,4): control = global bit12
  for (int t = wave * 2; t < wave * 2 + 2; ++t)
    wmma_group16(tile, M1, t * 16, 1, 256, x1, half, lm);
  __syncthreads();

  // G2: rows = bits 7-4 (stride 16), cols = (bits 11-8 fixed per tile) + bits 3-0
  const float2* M2 = gm + (s * 4 + 2) * 256;
  for (int t = wave * 2; t < wave * 2 + 2; ++t)
    wmma_group16(tile, M2, t * 256, 1, 16, (t & 1) ? 8 : 0, half, lm);
  __syncthreads();

  // G3: rows = bits 3-0 (stride 1), cols = bits 11-4 split by bit4 parity
  const float2* M3 = gm + (s * 4 + 3) * 256;
  for (int t = wave * 2; t < wave * 2 + 2; ++t) {
    int cb = t >> 1, pp = t & 1;
    wmma_group16(tile, M3, cb * 512 + pp * 16, 32, 1, pp ? 8 : 0, half, lm);
  }
  __syncthreads();

  // Async DMA: LDS -> global. S_ENDPGM's implicit wait-idle (and the explicit
  // s_wait_asynccnt below) guarantee completion before the next kernel.
  for (int e = threadIdx.x * 2; e < 4096; e += 512) {
    unsigned la = lds_addr32(&tile[e]);
    const float2* ga = psi + gbase + e;
    asm volatile("global_store_async_from_lds_b128 %0, %1, off"
                 :: "v"(ga), "v"(la) : "memory");
  }

  if (donorm) {
    float loc = 0.f;
    for (int i = threadIdx.x; i < 4096; i += 256) {
      float2 v = tile[i];
      loc += v.x * v.x + v.y * v.y;
    }
    red[threadIdx.x] = loc;
    __syncthreads();
    for (int off = 128; off; off >>= 1) {
      if (threadIdx.x < off) red[threadIdx.x] += red[threadIdx.x + off];
      __syncthreads();
    }
    if (threadIdx.x == 0) norms[(s * BATCH + b) * 16 + hi] = red[0];
  }
  asm volatile("s_wait_asynccnt 0" ::: "memory");
}

// ---------------------------------------------------------------------------
// Final cleanup: last pending CNOT(15,0) (swap across bit15 when bit0==1)
// + last normalization. Pairwise in one thread -> in-place safe.
// ---------------------------------------------------------------------------
__global__ __launch_bounds__(256) void k_final(float2* __restrict__ psi,
                                               const float* __restrict__ norms) {
  const int t = blockIdx.x * 256 + threadIdx.x;   // 128*32768 threads
  const int b = t >> 15;
  const int j = t & 32767;
  const float* np = norms + (7 * BATCH + b) * 16;
  float ss = 0.f;
  for (int i = 0; i < 16; ++i) ss += np[i];
  const float inv = 1.0f / (sqrtf(ss) + 1e-8f);
  const size_t i0 = ((size_t)b << 16) | (size_t)j;
  const size_t i1 = i0 | 32768u;
  float2 a = psi[i0], c = psi[i1];
  if (j & 1) { float2 tmp = a; a = c; c = tmp; }
  psi[i0] = make_float2(a.x * inv, a.y * inv);
  psi[i1] = make_float2(c.x * inv, c.y * inv);
}

// ---------------------------------------------------------------------------
extern "C" void kernel_launch(void* const* d_in, const int* in_sizes, int n_in,
                              void* d_out, int out_size, void* d_ws, size_t ws_size,
                              hipStream_t stream) {
  const float* psi0_re = (const float*)d_in[0];
  const float* psi0_im = (const float*)d_in[1];
  const float* W1 = (const float*)d_in[2];
  const float* b1 = (const float*)d_in[3];
  const float* W2 = (const float*)d_in[4];
  const float* b2 = (const float*)d_in[5];
  const float* W3 = (const float*)d_in[6];
  const float* b3 = (const float*)d_in[7];

  float2* psi = (float2*)d_out;                       // 128 x 65536 complex64
  float2* gm = (float2*)d_ws;                         // 8 steps x 4 groups x 256 = 64KB
  float* norms = (float*)d_ws + 2 * 8 * 4 * 256;      // 8 x 128 x 16 partials = 64KB

  k_init<<<(BATCH * QDIM) / 256, 256, 0, stream>>>(psi0_re, psi0_im, psi);
  k_net<<<8, HID, 0, stream>>>(W1, b1, W2, b2, W3, b3, gm);

  for (int s = 0; s < 8; ++s) {
    // layer 1: fold previous step's CNOT(15,0) (s>0) and its normalization
    k_high<<<4096, 256, 0, stream>>>(psi, gm, norms, s, (s > 0) ? 1 : 0, s - 1);
    k_low<<<2048, 256, 0, stream>>>(psi, gm, norms, s, 0);
    // layer 2: fold layer 1's CNOT(15,0)
    k_high<<<4096, 256, 0, stream>>>(psi, gm, norms, s, 1, -1);
    k_low<<<2048, 256, 0, stream>>>(psi, gm, norms, s, 1);
  }
  k_final<<<16384, 256, 0, stream>>>(psi, norms);
}